// L_Reg_47278999994676
// MI455X (gfx1250) — compile-verified
//
#include <hip/hip_runtime.h>
#include <hip/hip_bf16.h>
#include <stdint.h>

typedef float v2f __attribute__((ext_vector_type(2)));
typedef float v8f __attribute__((ext_vector_type(8)));
typedef unsigned int v4u __attribute__((ext_vector_type(4)));
typedef int v8i __attribute__((ext_vector_type(8)));
typedef int v4i __attribute__((ext_vector_type(4)));

#define CH_N     65536      // 256*256 elements per (B,C) channel
#define NCH      1024       // 16*64 channels
#define NTHREADS 256
#define NWAVES   8
#define KSEL     983        // int(65536 * 1.5 / 100)
#define NBINS    4096

// ---- dynamic LDS layout (bytes) ----
#define OFF_DATA 0
#define OFF_CNT  (CH_N * 4)               // 262144 : 4096 u32 histogram
#define OFF_BLK  (OFF_CNT + NBINS * 4)    // 278528 : 256 u32 block sums
#define OFF_SCAL (OFF_BLK + 256 * 4)      // 279552 : scalars
#define OFF_WP   (OFF_SCAL + 64)          // 279616 : 16 f32 wave partials
#define LDS_BYTES (OFF_WP + 64)           // 279680 (< 320 KB)

__device__ __forceinline__ unsigned order_key(float v) {
    unsigned u = __float_as_uint(v);
    unsigned m = (u & 0x80000000u) ? 0xFFFFFFFFu : 0x80000000u;
    return u ^ m; // monotone ascending uint key
}

__global__ __launch_bounds__(NTHREADS)
void topk_mse_channel_kernel(const float* __restrict__ x, float* __restrict__ partial)
{
    extern __shared__ unsigned char smem[];
    float*    dataF = (float*)(smem + OFF_DATA);
    unsigned* cntU  = (unsigned*)(smem + OFF_CNT);
    unsigned* blkU  = (unsigned*)(smem + OFF_BLK);
    unsigned* scalU = (unsigned*)(smem + OFF_SCAL);
    float*    scalF = (float*)(smem + OFF_SCAL);
    float*    wp    = (float*)(smem + OFF_WP);

    const unsigned tid  = threadIdx.x;
    const unsigned wave = tid >> 5;
    const unsigned lane = tid & 31;
    const unsigned ch   = blockIdx.x;
    const float* src = x + (size_t)ch * CH_N;

    // zero first-pass histogram while the DMA is in flight
    for (unsigned i = tid; i < NBINS; i += NTHREADS) cntU[i] = 0u;

    // ---- TDM: DMA the whole 256KB channel global -> LDS (wave 0 issues) ----
    if (wave == 0) {
#if __has_builtin(__builtin_amdgcn_tensor_load_to_lds)
        uint64_t ga = (uint64_t)(uintptr_t)src;
        unsigned ldsBase = __builtin_amdgcn_groupstaticsize() + OFF_DATA;
        v4u g0;
        g0[0] = 1u;                                   // count=1 (valid descriptor)
        g0[1] = ldsBase;                              // lds_addr (bytes)
        g0[2] = (unsigned)(ga & 0xFFFFFFFFull);       // global_addr[31:0]
        g0[3] = ((unsigned)(ga >> 32) & 0x01FFFFFFu)  // global_addr[56:32]
                | 0x80000000u;                        // type=2 ("image")
        v8i g1;
        g1[0] = (int)(2u << 16);       // wg_mask=0, data_size=2 (4 bytes)
        g1[1] = (int)(16384u << 16);   // tensor_dim0[15:0]=16384 in bits[63:48]
        g1[2] = (int)(4u << 16);       // tensor_dim0 hi16=0, tensor_dim1 lo16=4
        g1[3] = (int)(16384u << 16);   // tensor_dim1 hi16=0, tile_dim0=16384
        g1[4] = (int)4;                // tile_dim1=4, tile_dim2=0
        g1[5] = (int)16384;            // tensor_dim0_stride lo32 = 16384
        g1[6] = 0;                     // stride hi16 = 0, tensor_dim1_stride lo16 = 0
        g1[7] = 0;
        v4i gz4 = {0, 0, 0, 0};
        v8i gz8 = {0, 0, 0, 0, 0, 0, 0, 0};
        // 6-arg flavor on this toolchain: (g0, g1, g2, g3, g4, cpol)
        __builtin_amdgcn_tensor_load_to_lds(g0, g1, gz4, gz4, gz8, 0);
        __builtin_amdgcn_s_wait_tensorcnt(0);
#else
        for (unsigned i = lane; i < CH_N; i += 32) dataF[i] = src[i];
#endif
    }
    __syncthreads();

    // ---- Pass 1: top-12-bit histogram + channel sum / sumsq (WMMA reduction) ----
    float fsum = 0.f, fsq = 0.f;
#if __has_builtin(__builtin_amdgcn_wmma_f32_16x16x4_f32)
    v8f csum = {}; v8f csq = {};
    const v2f bones = {1.0f, 1.0f};   // B (4x16) = all ones -> D row-sums A
#endif
    for (unsigned it = 0; it < CH_N / (NWAVES * 64); ++it) {
        unsigned base = (it * NWAVES + wave) * 64 + lane * 2;
        v2f v = *(const v2f*)(dataF + base);
        atomicAdd(&cntU[order_key(v.x) >> 20], 1u);
        atomicAdd(&cntU[order_key(v.y) >> 20], 1u);
#if __has_builtin(__builtin_amdgcn_wmma_f32_16x16x4_f32)
        v2f vv = { v.x * v.x, v.y * v.y };
        csum = __builtin_amdgcn_wmma_f32_16x16x4_f32(false, v,  false, bones,
                                                     (short)0, csum, false, false);
        csq  = __builtin_amdgcn_wmma_f32_16x16x4_f32(false, vv, false, bones,
                                                     (short)0, csq,  false, false);
#else
        fsum += v.x + v.y;
        fsq  += v.x * v.x + v.y * v.y;
#endif
    }
#if __has_builtin(__builtin_amdgcn_wmma_f32_16x16x4_f32)
    for (int j = 0; j < 8; ++j) { fsum += csum[j]; fsq += csq[j]; }
#endif
    for (int off = 16; off; off >>= 1) {
        fsum += __shfl_xor(fsum, off, 32);
        fsq  += __shfl_xor(fsq,  off, 32);
    }
#if __has_builtin(__builtin_amdgcn_wmma_f32_16x16x4_f32)
    fsum *= (1.0f / 16.0f);  // D has 16 identical columns
    fsq  *= (1.0f / 16.0f);
#endif
    if (lane == 0) { wp[wave] = fsum; wp[8 + wave] = fsq; }
    __syncthreads();

    // ---- find boundary bin b1 (suffix scan from top) ----
    { unsigned s = 0, b0 = tid * 16;
      for (int j = 0; j < 16; ++j) s += cntU[b0 + j];
      blkU[tid] = s; }
    __syncthreads();
    if (tid == 0) {
        float S = 0.f, Q = 0.f;
        for (int wv = 0; wv < 8; ++wv) { S += wp[wv]; Q += wp[8 + wv]; }
        scalF[4] = S; scalF[5] = Q;
        unsigned need = KSEL; int blk = 255;
        while (blk > 0 && blkU[blk] < need) { need -= blkU[blk]; --blk; }
        int b = blk * 16 + 15;
        while (b > 0 && cntU[b] < need) { need -= cntU[b]; --b; }
        scalU[0] = (unsigned)b;   // b1
        scalU[1] = need;          // k' needed from bin b1 (>=1)
    }
    __syncthreads();
    const unsigned b1 = scalU[0];
    const unsigned kprime = scalU[1];
    __syncthreads();

    // ---- Pass 2: sub-histogram on key bits [19:8] within bin b1 ----
    for (unsigned i = tid; i < NBINS; i += NTHREADS) cntU[i] = 0u;
    __syncthreads();
    for (unsigned i = tid; i < CH_N; i += NTHREADS) {
        unsigned key = order_key(dataF[i]);
        if ((key >> 20) == b1) atomicAdd(&cntU[(key >> 8) & 0xFFFu], 1u);
    }
    __syncthreads();
    { unsigned s = 0, b0 = tid * 16;
      for (int j = 0; j < 16; ++j) s += cntU[b0 + j];
      blkU[tid] = s; }
    __syncthreads();
    if (tid == 0) {
        unsigned need = kprime; int blk = 255;
        while (blk > 0 && blkU[blk] < need) { need -= blkU[blk]; --blk; }
        int b = blk * 16 + 15;
        while (b > 0 && cntU[b] < need) { need -= cntU[b]; --b; }
        scalU[2] = (unsigned)b;   // b2
        scalU[3] = need;          // r taken from boundary sub-bin
        scalU[6] = cntU[b];       // candidate count in boundary sub-bin
    }
    __syncthreads();
    const unsigned thresh = (b1 << 12) | scalU[2];   // 24-bit key threshold
    const float r   = (float)scalU[3];
    const float ceq = (float)scalU[6];

    // ---- Pass 3: deterministic sums above / at the 24-bit threshold ----
    float sAbove = 0.f, sEq = 0.f;
    for (unsigned i = tid; i < CH_N; i += NTHREADS) {
        float xv = dataF[i];
        unsigned k24 = order_key(xv) >> 8;
        sAbove += (k24 > thresh)  ? xv : 0.f;
        sEq    += (k24 == thresh) ? xv : 0.f;
    }
    for (int off = 16; off; off >>= 1) {
        sAbove += __shfl_xor(sAbove, off, 32);
        sEq    += __shfl_xor(sEq,    off, 32);
    }
    if (lane == 0) { wp[wave] = sAbove; wp[8 + wave] = sEq; }
    __syncthreads();
    if (tid == 0) {
        float sa = 0.f, se = 0.f;
        for (int wv = 0; wv < 8; ++wv) { sa += wp[wv]; se += wp[8 + wv]; }
        float topsum = sa + ((ceq > 0.f) ? r * (se / ceq) : 0.f);
        float A = topsum / (float)KSEL;
        float S = scalF[4], Q = scalF[5];
        // sum over channel of (x - A)^2
        partial[ch] = Q - 2.0f * A * S + (float)CH_N * A * A;
    }
}

__global__ __launch_bounds__(256)
void topk_mse_reduce_kernel(const float* __restrict__ partial, float* __restrict__ out)
{
    __shared__ float red[256];
    const unsigned tid = threadIdx.x;
    float s = 0.f;
    for (int j = 0; j < NCH / 256; ++j) s += partial[tid * (NCH / 256) + j];
    red[tid] = s;
    __syncthreads();
    for (int st = 128; st; st >>= 1) {
        if (tid < (unsigned)st) red[tid] += red[tid + st];
        __syncthreads();
    }
    if (tid == 0) out[0] = red[0] * (1.0f / ((float)NCH * (float)CH_N));
}

extern "C" void kernel_launch(void* const* d_in, const int* in_sizes, int n_in,
                              void* d_out, int out_size, void* d_ws, size_t ws_size,
                              hipStream_t stream) {
    const float* x = (const float*)d_in[0];
    float* partial = (float*)d_ws;           // NCH floats of scratch
    float* out     = (float*)d_out;          // scalar fp32 result

    topk_mse_channel_kernel<<<NCH, NTHREADS, LDS_BYTES, stream>>>(x, partial);
    topk_mse_reduce_kernel<<<1, 256, 0, stream>>>(partial, out);
}